// DecoderFeatureFuser3D_72894184948158
// MI455X (gfx1250) — compile-verified
//
#include <hip/hip_runtime.h>
#include <stdint.h>

typedef __attribute__((ext_vector_type(2))) float v2f;
typedef __attribute__((ext_vector_type(8))) float v8f;

#define NPTS   16384
#define NB     4
#define CI     256
#define CO     128
#define C2D    128
#define IMH    96
#define IMW    160
#define HWSZ   (IMH * IMW)
#define TILE_N 64               // 4 WMMA N-subtiles of 16
#define NEG_SLOPE 0.1f

// ---------------------------------------------------------------------------
// Pack W [128 x 256] (row-major) into WMMA f32 16x16x4 A-operand layout:
//   wsA[((mt*64 + kstep)*32 + lane)*2 + j] = W[mt*16 + (lane&15)][kstep*4 + 2*(lane>>4) + j]
// so the GEMM loop reads one coalesced b64 per lane per K-step.
// ---------------------------------------------------------------------------
__global__ __launch_bounds__(256) void pack_w_kernel(const float* __restrict__ W,
                                                     float* __restrict__ wsA) {
    int tid   = blockIdx.x * 256 + threadIdx.x;   // 16384 threads total
    int lane  = tid & 31;
    int rest  = tid >> 5;
    int kstep = rest & 63;
    int mt    = rest >> 6;
    int m = mt * 16 + (lane & 15);
    int k = kstep * 4 + 2 * (lane >> 4);
    v2f a;
    a.x = W[m * CI + k];
    a.y = W[m * CI + k + 1];
    *(v2f*)(wsA + (size_t)((mt * 64 + kstep) * 32 + lane) * 2) = a;
}

// ---------------------------------------------------------------------------
// Fused: bilinear grid-sample -> concat -> 128x256 linear -> bias -> leakyReLU
// One block = one (batch, 64-point tile). 8 waves; wave w owns channel tile
// M = [16w, 16w+16) for all four 16-point N-subtiles (4 accumulators).
// ---------------------------------------------------------------------------
__global__ __launch_bounds__(256)
void fuse_kernel(const float* __restrict__ xy,      // [B,2,N]
                 const float* __restrict__ feat2d,  // [B,128,H,W]
                 const float* __restrict__ feat3d,  // [B,128,N]
                 const float* __restrict__ wsA,     // packed A operand
                 const float* __restrict__ bias,    // [128]
                 float* __restrict__ out) {         // [B,128,N]
    // B-operand staged in LDS, pre-swizzled for V_WMMA_F32_16X16X4_F32:
    // value (K, p):  nt=p>>4, kstep=K>>2, lane=((K>>1)&1)*16 + (p&15), j=K&1
    __shared__ float ldsB[4 * 64 * 32 * 2];   // 64 KB

    int blk = blockIdx.x;
    int b   = blk >> 8;              // NPTS/TILE_N = 256 tiles per batch
    int n0  = (blk & 255) * TILE_N;
    int t   = threadIdx.x;

    // ---------------- fill phase: sample + concat into LDS ----------------
    {
        int p = t & 63;            // point within tile (wave-contiguous)
        int g = t >> 6;            // channel group: channels [32g, 32g+32)
        int n = n0 + p;

        int nt      = p >> 4;
        int lane_lo = p & 15;
        uint32_t lds_base = (uint32_t)(uintptr_t)(&ldsB[0]); // low 32b = LDS offset

        // --- feat_3d half of the concat: async DMA straight into the
        //     swizzled B-operand slot in LDS (gfx1250 async-to-LDS path).
        const float* f3 = feat3d + (size_t)b * C2D * NPTS + n;
        #pragma unroll
        for (int cc = 0; cc < 32; ++cc) {
            int c  = g * 32 + cc;
            int K3 = 128 + c;
            uint32_t loff = lds_base +
                (uint32_t)((((nt * 64 + (K3 >> 2)) * 32 +
                             ((K3 >> 1) & 1) * 16 + lane_lo) * 2 + (K3 & 1)) * 4);
            uint64_t gaddr = (uint64_t)(uintptr_t)(f3 + (size_t)c * NPTS);
            asm volatile("global_load_async_to_lds_b32 %0, %1, off"
                         :: "v"(loff), "v"(gaddr) : "memory");
        }

        // --- bilinear-sampled 2D half: gather + lerp in VALU, ds_store.
        float x = xy[((size_t)b * 2 + 0) * NPTS + n];
        float y = xy[((size_t)b * 2 + 1) * NPTS + n];
        float x0f = floorf(x), y0f = floorf(y);
        int   ix0 = (int)x0f,  iy0 = (int)y0f;
        float wx1 = x - x0f,   wy1 = y - y0f;
        float wx0 = 1.0f - wx1, wy0 = 1.0f - wy1;
        int ix1 = ix0 + 1, iy1 = iy0 + 1;
        bool vx0 = (ix0 >= 0) && (ix0 < IMW), vx1 = (ix1 >= 0) && (ix1 < IMW);
        bool vy0 = (iy0 >= 0) && (iy0 < IMH), vy1 = (iy1 >= 0) && (iy1 < IMH);
        int cx0 = min(max(ix0, 0), IMW - 1), cx1 = min(max(ix1, 0), IMW - 1);
        int cy0 = min(max(iy0, 0), IMH - 1), cy1 = min(max(iy1, 0), IMH - 1);
        float w00 = (vx0 && vy0) ? wx0 * wy0 : 0.0f;
        float w10 = (vx1 && vy0) ? wx1 * wy0 : 0.0f;
        float w01 = (vx0 && vy1) ? wx0 * wy1 : 0.0f;
        float w11 = (vx1 && vy1) ? wx1 * wy1 : 0.0f;

        const float* fb = feat2d + (size_t)b * C2D * HWSZ;
        int o00 = cy0 * IMW + cx0, o10 = cy0 * IMW + cx1;
        int o01 = cy1 * IMW + cx0, o11 = cy1 * IMW + cx1;

        #pragma unroll 8
        for (int cc = 0; cc < 32; ++cc) {
            int c = g * 32 + cc;
            const float* fc = fb + (size_t)c * HWSZ;
            float val = w00 * fc[o00] + w10 * fc[o10] + w01 * fc[o01] + w11 * fc[o11];
            int K  = c;                          // sampled 2D features: K = 0..127
            int ln = ((K >> 1) & 1) * 16 + lane_lo;
            ldsB[((nt * 64 + (K >> 2)) * 32 + ln) * 2 + (K & 1)] = val;
        }

        // Wave-local async completion before signaling the barrier.
        asm volatile("s_wait_asynccnt 0" ::: "memory");
    }
    __syncthreads();

    // ---------------- GEMM phase: D = W_tile x cat_tile -------------------
    int lane = t & 31;
    int mt   = t >> 5;          // wave id == output-channel tile
    v8f acc[4] = {v8f{}, v8f{}, v8f{}, v8f{}};
    const float* wA = wsA + (size_t)mt * 64 * 32 * 2;

    #pragma unroll 4
    for (int ks = 0; ks < 64; ++ks) {
        v2f a  = *(const v2f*)(wA + (size_t)(ks * 32 + lane) * 2);
        v2f b0 = *(const v2f*)(ldsB + ((0 * 64 + ks) * 32 + lane) * 2);
        v2f b1 = *(const v2f*)(ldsB + ((1 * 64 + ks) * 32 + lane) * 2);
        v2f b2 = *(const v2f*)(ldsB + ((2 * 64 + ks) * 32 + lane) * 2);
        v2f b3 = *(const v2f*)(ldsB + ((3 * 64 + ks) * 32 + lane) * 2);
        acc[0] = __builtin_amdgcn_wmma_f32_16x16x4_f32(false, a, false, b0,
                                                       (short)0, acc[0], false, false);
        acc[1] = __builtin_amdgcn_wmma_f32_16x16x4_f32(false, a, false, b1,
                                                       (short)0, acc[1], false, false);
        acc[2] = __builtin_amdgcn_wmma_f32_16x16x4_f32(false, a, false, b2,
                                                       (short)0, acc[2], false, false);
        acc[3] = __builtin_amdgcn_wmma_f32_16x16x4_f32(false, a, false, b3,
                                                       (short)0, acc[3], false, false);
    }

    // ---------------- epilogue: bias + leaky ReLU + store -----------------
    // C/D layout: VGPR r, lane L -> M = r + 8*(L>>4), N = L&15
    int mrow_base = mt * 16 + 8 * (lane >> 4);
    int ncol      = lane & 15;
    #pragma unroll
    for (int r = 0; r < 8; ++r) {
        int ch   = mrow_base + r;
        float bv = bias[ch];
        size_t rowoff = ((size_t)b * CO + ch) * NPTS + n0 + ncol;
        #pragma unroll
        for (int nt = 0; nt < 4; ++nt) {
            float yv = acc[nt][r] + bv;
            yv = (yv >= 0.0f) ? yv : NEG_SLOPE * yv;
            out[rowoff + nt * 16] = yv;
        }
    }
}

// ---------------------------------------------------------------------------
extern "C" void kernel_launch(void* const* d_in, const int* in_sizes, int n_in,
                              void* d_out, int out_size, void* d_ws, size_t ws_size,
                              hipStream_t stream) {
    const float* xy     = (const float*)d_in[0];   // [4,2,16384]
    const float* feat2d = (const float*)d_in[1];   // [4,128,96,160]
    const float* feat3d = (const float*)d_in[2];   // [4,128,16384]
    const float* W      = (const float*)d_in[3];   // [128,256]
    const float* bias   = (const float*)d_in[4];   // [128]
    float*       out    = (float*)d_out;           // [4,128,16384]
    float*       wsA    = (float*)d_ws;            // needs 128 KB

    // Pack W into WMMA A-operand layout (deterministic, every call).
    pack_w_kernel<<<64, 256, 0, stream>>>(W, wsA);

    // One block per (batch, 64-point tile): 4 * 256 = 1024 blocks.
    fuse_kernel<<<NB * (NPTS / TILE_N), 256, 0, stream>>>(xy, feat2d, feat3d,
                                                          wsA, bias, out);
}